// DirectionalSAGEConv_19610820673957
// MI455X (gfx1250) — compile-verified
//
#include <hip/hip_runtime.h>
#include <hip/hip_bf16.h>

#define N_NODES 50000
#define N_EDGES 800000
#define CH 128            // IN_CH == OUT_CH == 128

typedef __attribute__((ext_vector_type(2))) float v2f;
typedef __attribute__((ext_vector_type(8))) float v8f;

// ---------------------------------------------------------------------------
// Kernel 1: zero the accumulator + count workspace (graph-safe, no memset API)
// ---------------------------------------------------------------------------
__global__ void sage_zero_ws(float* __restrict__ agg, float* __restrict__ cnt) {
    int i = blockIdx.x * blockDim.x + threadIdx.x;
    if (i < N_NODES * CH) agg[i] = 0.0f;
    if (i < N_NODES)      cnt[i] = 0.0f;
}

// ---------------------------------------------------------------------------
// Kernel 2: edge scatter-add. One wave32 per edge; lane l handles channels
// 4l..4l+3 (float4 gather from x[src], 4 global f32 atomics into agg[dst]).
// agg (25.6 MB) is L2-resident on MI455X (192 MB L2), so atomics resolve in L2.
// ---------------------------------------------------------------------------
__global__ void sage_scatter(const long long* __restrict__ ei,
                             const float* __restrict__ x,
                             float* __restrict__ agg,
                             float* __restrict__ cnt) {
    const int wavesPerBlock = blockDim.x >> 5;
    const int e = blockIdx.x * wavesPerBlock + (threadIdx.x >> 5);
    if (e >= N_EDGES) return;
    const int lane = threadIdx.x & 31;

    const long long src = ei[e];            // row 0 of edge_index
    const long long dst = ei[N_EDGES + e];  // row 1 of edge_index

    const float4 v = *(const float4*)(x + (size_t)src * CH + lane * 4);
    float* base = agg + (size_t)dst * CH + lane * 4;
    unsafeAtomicAdd(base + 0, v.x);
    unsafeAtomicAdd(base + 1, v.y);
    unsafeAtomicAdd(base + 2, v.z);
    unsafeAtomicAdd(base + 3, v.w);
    if (lane == 0) unsafeAtomicAdd(cnt + dst, 1.0f);
}

// ---------------------------------------------------------------------------
// Kernel 3: in-place mean  agg[i,c] /= max(cnt[i], 1)
// ---------------------------------------------------------------------------
__global__ void sage_mean(float* __restrict__ agg, const float* __restrict__ cnt) {
    int i = blockIdx.x * blockDim.x + threadIdx.x;
    if (i < N_NODES * CH) {
        float c = cnt[i >> 7];               // CH == 128
        agg[i] = agg[i] / fmaxf(c, 1.0f);
    }
}

// ---------------------------------------------------------------------------
// Kernel 4: fused dual GEMM + bias + ReLU using full-precision f32 WMMA:
//   out = relu( mean @ W_l^T + b_l + x @ W_r^T )
// Block = 256 threads = 8 waves; block handles a 16-row node tile, each wave
// one 16-col output tile. Both GEMMs accumulate into one v8f accumulator via
// V_WMMA_F32_16X16X4_F32 (K stepped by 4, 64 WMMA ops per wave).
//
// f32 16x16x4 fragment layout (ISA 7.12.2):
//   A (16x4): lanes 0-15 -> M=lane, V0=K0,V1=K1 ; lanes 16-31 -> V0=K2,V1=K3
//   B (4x16): lanes 0-15 -> N=lane, V0=K0,V1=K1 ; lanes 16-31 -> V0=K2,V1=K3
//   D (16x16): VGPR v -> M = v + 8*half, N = lane%16
// ---------------------------------------------------------------------------
__global__ void sage_wmma_gemm(const float* __restrict__ mean,
                               const float* __restrict__ x,
                               const float* __restrict__ Wl,
                               const float* __restrict__ bl,
                               const float* __restrict__ Wr,
                               float* __restrict__ out) {
    const int lane  = threadIdx.x & 31;
    const int wave  = threadIdx.x >> 5;     // 0..7: output column tile
    const int itile = blockIdx.x;           // 0..3124: node row tile (50000 = 3125*16)
    const int half  = lane >> 4;            // 0 or 1 (selects K pair)
    const int r     = lane & 15;

    const size_t row = (size_t)itile * 16 + r;   // A-fragment row (node)
    const size_t col = (size_t)wave * 16 + r;    // B-fragment / D column (channel)

    const float* mrow = mean + row * CH;
    const float* xrow = x    + row * CH;
    const float* wlr  = Wl   + col * CH;    // W_l[col][k] == (W_l^T)[k][col]
    const float* wrr  = Wr   + col * CH;

    v8f acc = {0.f, 0.f, 0.f, 0.f, 0.f, 0.f, 0.f, 0.f};

#pragma unroll
    for (int k0 = 0; k0 < CH; k0 += 4) {
        const int kk = k0 + 2 * half;
        v2f a0 = *(const v2f*)(mrow + kk);
        v2f b0 = *(const v2f*)(wlr + kk);
        acc = __builtin_amdgcn_wmma_f32_16x16x4_f32(
                  false, a0, false, b0, (short)0, acc, false, false);
        v2f a1 = *(const v2f*)(xrow + kk);
        v2f b1 = *(const v2f*)(wrr + kk);
        acc = __builtin_amdgcn_wmma_f32_16x16x4_f32(
                  false, a1, false, b1, (short)0, acc, false, false);
    }

    const float bias = bl[col];
#pragma unroll
    for (int v = 0; v < 8; ++v) {
        const size_t m = (size_t)itile * 16 + v + 8 * half;  // D-row for VGPR v
        float val = acc[v] + bias;
        out[m * CH + col] = val > 0.0f ? val : 0.0f;
    }
}

// ---------------------------------------------------------------------------
extern "C" void kernel_launch(void* const* d_in, const int* in_sizes, int n_in,
                              void* d_out, int out_size, void* d_ws, size_t ws_size,
                              hipStream_t stream) {
    const float*     x   = (const float*)d_in[0];
    const long long* ei  = (const long long*)d_in[1];  // int64 [2, N_EDGES]
    const float*     Wl  = (const float*)d_in[2];
    const float*     bl  = (const float*)d_in[3];
    const float*     Wr  = (const float*)d_in[4];
    float*           out = (float*)d_out;

    float* agg = (float*)d_ws;                 // N_NODES*CH floats (25.6 MB)
    float* cnt = agg + (size_t)N_NODES * CH;   // N_NODES floats

    // 1) zero workspace
    {
        int total = N_NODES * CH;
        int grid = (total + 255) / 256;
        sage_zero_ws<<<grid, 256, 0, stream>>>(agg, cnt);
    }
    // 2) scatter-add over edges: 8 waves/block -> 8 edges/block
    {
        int edgesPerBlock = 256 / 32;
        int grid = (N_EDGES + edgesPerBlock - 1) / edgesPerBlock;
        sage_scatter<<<grid, 256, 0, stream>>>(ei, x, agg, cnt);
    }
    // 3) mean (in place)
    {
        int total = N_NODES * CH;
        int grid = (total + 255) / 256;
        sage_mean<<<grid, 256, 0, stream>>>(agg, cnt);
    }
    // 4) fused dual-GEMM + bias + ReLU with f32 WMMA
    {
        int grid = N_NODES / 16;  // 3125 tiles, exact
        sage_wmma_gemm<<<grid, 256, 0, stream>>>(agg, x, Wl, bl, Wr, out);
    }
}